// L1OutUB_59785944760628
// MI455X (gfx1250) — compile-verified
//
#include <hip/hip_runtime.h>
#include <cmath>

// ---------------------------------------------------------------------------
// Problem constants (match the reference)
// ---------------------------------------------------------------------------
#define BB 512   // batch
#define DD 768   // x/y dim
#define HH 384   // hidden

typedef __attribute__((ext_vector_type(2))) float v2f;
typedef __attribute__((ext_vector_type(8))) float v8f;

#define MODE_RELU 0
#define MODE_NONE 1
#define MODE_TANH 2

// ---------------------------------------------------------------------------
// FP32 WMMA GEMM: C = act(A[MxK] @ B[KxN] + bias[N])
// One wave32 per 16(M) x 64(N) output strip, 4 x v8f accumulators,
// V_WMMA_F32_16X16X4_F32 along K (A fragment reused by 4 WMMAs).
// K / N / MODE are compile-time so the backend folds the B row stride into
// immediate load offsets and strength-reduces the K-loop to pointer bumps.
//
// Fragment layouts per CDNA5 ISA 7.12.2:
//   A (16x4 f32): lane r=lane&15 -> row M=r; half=lane>>4 selects K pair:
//                 v2f = { A[r, k+2*half], A[r, k+2*half+1] }
//   B (4x16 f32): col = lane&15;  v2f = { B[k+2*half, col], B[k+2*half+1, col] }
//   C/D (16x16 f32): VGPR v, lanes 0-15 -> M=v, lanes 16-31 -> M=v+8; N=lane&15
// ---------------------------------------------------------------------------
template<int K, int N, int MODE>
__global__ __launch_bounds__(32)
void gemm16x64_f32_wmma(const float* __restrict__ A,
                        const float* __restrict__ Bm,
                        const float* __restrict__ bias,
                        float* __restrict__ C,
                        float* __restrict__ C2)   // second output (iv) for MODE_TANH
{
    const int lane = threadIdx.x;        // 0..31, wave32
    const int r    = lane & 15;
    const int half = lane >> 4;          // 0 or 1 (K-half / M-half select)
    const int tileM = blockIdx.y << 4;   // 16 rows per wave
    const int tileN = blockIdx.x << 6;   // 64 cols per wave

    v8f acc0 = {}; v8f acc1 = {}; v8f acc2 = {}; v8f acc3 = {};

    const int cb = tileN + r;
    // Per-lane base pointers; the K loop advances these by constants only.
    const float* __restrict__ pA = A + (size_t)(tileM + r) * K + 2 * half;
    const float* __restrict__ pB = Bm + (size_t)(2 * half) * N + cb;

#pragma unroll 4
    for (int k = 0; k < K; k += 4) {
        // A fragment (shared by the 4 WMMAs of this K-step)
        v2f a;
        a.x = pA[0];
        a.y = pA[1];

        // speculative prefetch of the A row ahead (lowers to global_prefetch_b8)
        __builtin_prefetch(pA + 32, 0, 0);

        v2f b0; b0.x = pB[0];  b0.y = pB[N];
        v2f b1; b1.x = pB[16]; b1.y = pB[N + 16];
        v2f b2; b2.x = pB[32]; b2.y = pB[N + 32];
        v2f b3; b3.x = pB[48]; b3.y = pB[N + 48];

        acc0 = __builtin_amdgcn_wmma_f32_16x16x4_f32(false, a, false, b0, (short)0, acc0, false, false);
        acc1 = __builtin_amdgcn_wmma_f32_16x16x4_f32(false, a, false, b1, (short)0, acc1, false, false);
        acc2 = __builtin_amdgcn_wmma_f32_16x16x4_f32(false, a, false, b2, (short)0, acc2, false, false);
        acc3 = __builtin_amdgcn_wmma_f32_16x16x4_f32(false, a, false, b3, (short)0, acc3, false, false);

        pA += 4;
        pB += 4 * N;
    }

    const float bv0 = bias[cb];
    const float bv1 = bias[cb + 16];
    const float bv2 = bias[cb + 32];
    const float bv3 = bias[cb + 48];

#pragma unroll
    for (int v = 0; v < 8; ++v) {
        const int row = tileM + v + 8 * half;
        const size_t base = (size_t)row * N;
        float e0 = acc0[v] + bv0;
        float e1 = acc1[v] + bv1;
        float e2 = acc2[v] + bv2;
        float e3 = acc3[v] + bv3;
        if (MODE == MODE_RELU) {
            C[base + cb]      = fmaxf(e0, 0.0f);
            C[base + cb + 16] = fmaxf(e1, 0.0f);
            C[base + cb + 32] = fmaxf(e2, 0.0f);
            C[base + cb + 48] = fmaxf(e3, 0.0f);
        } else if (MODE == MODE_TANH) {
            float t0 = tanhf(e0), t1 = tanhf(e1), t2 = tanhf(e2), t3 = tanhf(e3);
            C[base + cb]      = t0;  C2[base + cb]      = expf(-t0);
            C[base + cb + 16] = t1;  C2[base + cb + 16] = expf(-t1);
            C[base + cb + 32] = t2;  C2[base + cb + 32] = expf(-t2);
            C[base + cb + 48] = t3;  C2[base + cb + 48] = expf(-t3);
        } else { // MODE_NONE
            C[base + cb]      = e0;
            C[base + cb + 16] = e1;
            C[base + cb + 32] = e2;
            C[base + cb + 48] = e3;
        }
    }
}

// ---------------------------------------------------------------------------
// Column sums of y: Ybar[d] = sum_j y[j,d], Y2[d] = sum_j y[j,d]^2
// (deterministic: one thread per column, fixed-order loop)
// ---------------------------------------------------------------------------
__global__ void colsums_kernel(const float* __restrict__ y,
                               float* __restrict__ ybar,
                               float* __restrict__ y2)
{
    const int d = blockIdx.x * blockDim.x + threadIdx.x;
    if (d >= DD) return;
    float s = 0.0f, s2 = 0.0f;
    for (int j = 0; j < BB; ++j) {
        const float v = y[(size_t)j * DD + d];
        s  += v;
        s2 += v * v;
    }
    ybar[d] = s;
    y2[d]   = s2;
}

// ---------------------------------------------------------------------------
// Per-row reductions. For row i:
//   Ppart[i] = sum_d [ -0.5*(mu-y)^2*iv - 0.5*logvar ]            (positive_i)
//   Spart[i] = sum_j all_probs[i,j]  (collapsed analytically)
//            = -0.5*( dot(Y2,iv_i) - 2*dot(Ybar, mu_i*iv_i) + B*dot(mu_i^2,iv_i) )
//              - 0.5*B*sum_d logvar_i
// Deterministic LDS tree reduction.
// ---------------------------------------------------------------------------
__global__ __launch_bounds__(256)
void rowreduce_kernel(const float* __restrict__ mu,
                      const float* __restrict__ logvar,
                      const float* __restrict__ iv,
                      const float* __restrict__ y,
                      const float* __restrict__ ybar,
                      const float* __restrict__ y2,
                      float* __restrict__ Ppart,
                      float* __restrict__ Spart)
{
    const int i = blockIdx.x;
    const size_t rowoff = (size_t)i * DD;

    float p = 0.0f, t1 = 0.0f, t2 = 0.0f, t3 = 0.0f, t4 = 0.0f;
    for (int d = threadIdx.x; d < DD; d += blockDim.x) {
        const float m  = mu[rowoff + d];
        const float lv = logvar[rowoff + d];
        const float v  = iv[rowoff + d];
        const float yy = y[rowoff + d];
        const float df = m - yy;
        p  += -0.5f * df * df * v - 0.5f * lv;
        t1 += y2[d] * v;
        t2 += ybar[d] * m * v;
        t3 += m * m * v;
        t4 += lv;
    }

    __shared__ float red[256];
    float vals[5] = { p, t1, t2, t3, t4 };
    float out[5];
#pragma unroll
    for (int q = 0; q < 5; ++q) {
        red[threadIdx.x] = vals[q];
        __syncthreads();
        for (int off = 128; off > 0; off >>= 1) {
            if ((int)threadIdx.x < off) red[threadIdx.x] += red[threadIdx.x + off];
            __syncthreads();
        }
        out[q] = red[0];
        __syncthreads();
    }

    if (threadIdx.x == 0) {
        Ppart[i] = out[0];
        Spart[i] = -0.5f * (out[1] - 2.0f * out[2] + (float)BB * out[3])
                   - 0.5f * (float)BB * out[4];
    }
}

// ---------------------------------------------------------------------------
// Final scalar:  out = P/B - S/B^2 - C,  C = log1p(exp(-20)/(B-1))
// (negative[i,j] = all_probs[i,j] + C exactly, from the degenerate logsumexp)
// ---------------------------------------------------------------------------
__global__ __launch_bounds__(256)
void finalize_kernel(const float* __restrict__ Ppart,
                     const float* __restrict__ Spart,
                     float* __restrict__ out)
{
    float p = 0.0f, s = 0.0f;
    for (int i = threadIdx.x; i < BB; i += blockDim.x) {
        p += Ppart[i];
        s += Spart[i];
    }
    __shared__ float redp[256];
    __shared__ float reds[256];
    redp[threadIdx.x] = p;
    reds[threadIdx.x] = s;
    __syncthreads();
    for (int off = 128; off > 0; off >>= 1) {
        if ((int)threadIdx.x < off) {
            redp[threadIdx.x] += redp[threadIdx.x + off];
            reds[threadIdx.x] += reds[threadIdx.x + off];
        }
        __syncthreads();
    }
    if (threadIdx.x == 0) {
        const float Bf = (float)BB;
        const float Cshift = log1pf(expf(-20.0f) / (Bf - 1.0f));
        out[0] = redp[0] / Bf - reds[0] / (Bf * Bf) - Cshift;
    }
}

// ---------------------------------------------------------------------------
// kernel_launch
// inputs: x, y, mu_w1, mu_b1, mu_w2, mu_b2, lv_w1, lv_b1, lv_w2, lv_b2 (all f32)
// ---------------------------------------------------------------------------
extern "C" void kernel_launch(void* const* d_in, const int* in_sizes, int n_in,
                              void* d_out, int out_size, void* d_ws, size_t ws_size,
                              hipStream_t stream)
{
    (void)in_sizes; (void)n_in; (void)out_size; (void)ws_size;

    const float* X     = (const float*)d_in[0];
    const float* Y     = (const float*)d_in[1];
    const float* mu_w1 = (const float*)d_in[2];
    const float* mu_b1 = (const float*)d_in[3];
    const float* mu_w2 = (const float*)d_in[4];
    const float* mu_b2 = (const float*)d_in[5];
    const float* lv_w1 = (const float*)d_in[6];
    const float* lv_b1 = (const float*)d_in[7];
    const float* lv_w2 = (const float*)d_in[8];
    const float* lv_b2 = (const float*)d_in[9];
    float* out = (float*)d_out;

    // Workspace layout (floats)
    float* ws    = (float*)d_ws;
    float* Hmu   = ws;                      // B*H
    float* Hlv   = Hmu   + (size_t)BB * HH; // B*H
    float* Mu    = Hlv   + (size_t)BB * HH; // B*D
    float* Lvar  = Mu    + (size_t)BB * DD; // B*D
    float* Iv    = Lvar  + (size_t)BB * DD; // B*D
    float* Ybar  = Iv    + (size_t)BB * DD; // D
    float* Y2    = Ybar  + DD;              // D
    float* Ppart = Y2    + DD;              // B
    float* Spart = Ppart + BB;              // B

    // Layer 1: H = relu(X @ W1 + b1), M=512 K=768 N=384
    {
        dim3 grid(HH / 64, BB / 16);
        gemm16x64_f32_wmma<DD, HH, MODE_RELU>
            <<<grid, 32, 0, stream>>>(X, mu_w1, mu_b1, Hmu, nullptr);
        gemm16x64_f32_wmma<DD, HH, MODE_RELU>
            <<<grid, 32, 0, stream>>>(X, lv_w1, lv_b1, Hlv, nullptr);
    }

    // Layer 2: mu = Hmu @ W2 + b2 ; logvar = tanh(Hlv @ W2 + b2), iv = exp(-logvar)
    // M=512 K=384 N=768
    {
        dim3 grid(DD / 64, BB / 16);
        gemm16x64_f32_wmma<HH, DD, MODE_NONE>
            <<<grid, 32, 0, stream>>>(Hmu, mu_w2, mu_b2, Mu, nullptr);
        gemm16x64_f32_wmma<HH, DD, MODE_TANH>
            <<<grid, 32, 0, stream>>>(Hlv, lv_w2, lv_b2, Lvar, Iv);
    }

    // Column sums of y
    colsums_kernel<<<(DD + 255) / 256, 256, 0, stream>>>(Y, Ybar, Y2);

    // Per-row reductions (positive_i and collapsed pairwise row sums)
    rowreduce_kernel<<<BB, 256, 0, stream>>>(Mu, Lvar, Iv, Y, Ybar, Y2, Ppart, Spart);

    // Final scalar
    finalize_kernel<<<1, 256, 0, stream>>>(Ppart, Spart, out);
}